// baseGNN_10608569221612
// MI455X (gfx1250) — compile-verified
//
#include <hip/hip_runtime.h>
#include <math.h>

#define F_NODE 128
#define NGRAPH 1024
#define NCLASS 4
#define STATS_BLOCKS 2048
#define BN_EPS 1e-5f

typedef __attribute__((ext_vector_type(2))) float v2f;
typedef __attribute__((ext_vector_type(4))) float v4f;
typedef __attribute__((ext_vector_type(8))) float v8f;

// ---------------------------------------------------------------------------
// Kernel 1: per-block partial sum / sum-of-squares per feature (deterministic).
// 256 threads: lane fg (0..31) owns features 4*fg..4*fg+3 (one b128 load),
// ro (0..7) strides rows. A single wave covers a full 512B row per load burst.
// ---------------------------------------------------------------------------
__global__ void stats_partial(const float* __restrict__ x, int N,
                              float* __restrict__ psum, float* __restrict__ psq) {
    int tid = threadIdx.x;              // 0..255
    int fg  = tid & 31;                 // feature group: features 4*fg..4*fg+3
    int ro  = tid >> 5;                 // row offset 0..7
    int bid = blockIdx.x;
    int rpb = (N + STATS_BLOCKS - 1) / STATS_BLOCKS;
    int r0  = bid * rpb;
    int r1  = r0 + rpb; if (r1 > N) r1 = N;

    v4f s = {}, q = {};
    for (int r = r0 + ro; r < r1; r += 8) {
        v4f v = *(const v4f*)(x + (long)r * F_NODE + fg * 4);
        s += v;
        q += v * v;
    }

    __shared__ v4f sms[256];
    __shared__ v4f smq[256];
    sms[ro * 32 + fg] = s;
    smq[ro * 32 + fg] = q;
    __syncthreads();
    for (int st = 4; st > 0; st >>= 1) {
        if (ro < st) {
            sms[ro * 32 + fg] += sms[(ro + st) * 32 + fg];
            smq[ro * 32 + fg] += smq[(ro + st) * 32 + fg];
        }
        __syncthreads();
    }
    if (ro == 0) {
        *(v4f*)(psum + (long)bid * F_NODE + fg * 4) = sms[fg];
        *(v4f*)(psq  + (long)bid * F_NODE + fg * 4) = smq[fg];
    }
}

// ---------------------------------------------------------------------------
// Kernel 2: reduce partials per feature -> fused affine (scale, shift),
// x_norm = x*scale[f] + shift[f].
// ---------------------------------------------------------------------------
__global__ void stats_finalize(const float* __restrict__ psum,
                               const float* __restrict__ psq,
                               const float* __restrict__ gamma,
                               const float* __restrict__ beta,
                               int N,
                               float* __restrict__ scale,
                               float* __restrict__ shift) {
    int f   = blockIdx.x;               // one block per feature
    int tid = threadIdx.x;              // 256 threads
    __shared__ float ss[256];
    __shared__ float sq[256];
    float a = 0.f, b = 0.f;
    for (int p = tid; p < STATS_BLOCKS; p += 256) {
        a += psum[p * F_NODE + f];
        b += psq [p * F_NODE + f];
    }
    ss[tid] = a; sq[tid] = b;
    __syncthreads();
    for (int st = 128; st > 0; st >>= 1) {
        if (tid < st) { ss[tid] += ss[tid + st]; sq[tid] += sq[tid + st]; }
        __syncthreads();
    }
    if (tid == 0) {
        float invN = 1.0f / (float)N;
        float mu   = ss[0] * invN;
        float var  = sq[0] * invN - mu * mu;   // biased variance
        float rs   = rsqrtf(var + BN_EPS);
        float sc   = gamma[f] * rs;
        scale[f] = sc;
        shift[f] = beta[f] - mu * sc;
    }
}

// ---------------------------------------------------------------------------
// Segment boundaries from the sorted batch vector: graph g = [start[g], start[g+1])
// ---------------------------------------------------------------------------
__global__ void seg_init(int* __restrict__ start, int N) {
    int g = blockIdx.x * blockDim.x + threadIdx.x;
    if (g <= NGRAPH) start[g] = N;      // default: empty tail segments
}

__global__ void seg_bounds(const int* __restrict__ batch, int N,
                           int* __restrict__ start) {
    int i = blockIdx.x * blockDim.x + threadIdx.x;
    if (i >= N) return;
    int bi = batch[i];
    int bp = (i == 0) ? -1 : batch[i - 1];
    for (int g = bp + 1; g <= bi; ++g) start[g] = i;
}

// ---------------------------------------------------------------------------
// Kernel 3: per-graph softmax aggregation. One workgroup (256 thr = 8 waves)
// per graph. Lane fg owns 4 consecutive features (b128 loads), ro strides rows.
// Pass A: per-feature max of t*x_norm. Pass B re-reads the same ~500KB segment
// (L2-resident) accumulating exp-sum and exp-weighted sum.
// pooled[g,f] = sum(e*xn)/sum(e) == segment softmax aggregation.
// ---------------------------------------------------------------------------
__global__ void graph_pool(const float* __restrict__ x,
                           const int*  __restrict__ start,
                           const float* __restrict__ scale,
                           const float* __restrict__ shift,
                           const float* __restrict__ tptr,
                           float* __restrict__ pooled) {
    int g   = blockIdx.x;
    int tid = threadIdx.x;              // 256
    int fg  = tid & 31;                 // features 4*fg..4*fg+3
    int ro  = tid >> 5;                 // row offset 0..7
    int s = start[g];
    int e = start[g + 1];
    float t = *tptr;
    v4f sc = *(const v4f*)(scale + fg * 4);
    v4f sh = *(const v4f*)(shift + fg * 4);

    __shared__ v4f redA[256];
    __shared__ v4f redB[256];

    // Pass A: per-feature max of logits over the segment
    float ninf = -__builtin_inff();
    v4f lmax = { ninf, ninf, ninf, ninf };
    for (int r = s + ro; r < e; r += 8) {
        v4f xn = *(const v4f*)(x + (long)r * F_NODE + fg * 4) * sc + sh;
        lmax = __builtin_elementwise_max(lmax, xn * t);
    }
    redA[ro * 32 + fg] = lmax;
    __syncthreads();
    for (int st = 4; st > 0; st >>= 1) {
        if (ro < st)
            redA[ro * 32 + fg] =
                __builtin_elementwise_max(redA[ro * 32 + fg], redA[(ro + st) * 32 + fg]);
        __syncthreads();
    }
    v4f gmax = redA[fg];                // broadcast to all row groups
    __syncthreads();

    // Pass B: exp-sum and exp-weighted sum (segment re-read hits L2)
    v4f den = {}, num = {};
    for (int r = s + ro; r < e; r += 8) {
        v4f xn = *(const v4f*)(x + (long)r * F_NODE + fg * 4) * sc + sh;
        v4f ex;
        ex.x = __expf(t * xn.x - gmax.x);
        ex.y = __expf(t * xn.y - gmax.y);
        ex.z = __expf(t * xn.z - gmax.z);
        ex.w = __expf(t * xn.w - gmax.w);
        den += ex;
        num += ex * xn;
    }
    redA[ro * 32 + fg] = den;
    redB[ro * 32 + fg] = num;
    __syncthreads();
    for (int st = 4; st > 0; st >>= 1) {
        if (ro < st) {
            redA[ro * 32 + fg] += redA[(ro + st) * 32 + fg];
            redB[ro * 32 + fg] += redB[(ro + st) * 32 + fg];
        }
        __syncthreads();
    }
    if (ro == 0) {
        v4f out = redB[fg] / redA[fg];
        *(v4f*)(pooled + (long)g * F_NODE + fg * 4) = out;
    }
}

// ---------------------------------------------------------------------------
// Kernel 4: final projection pooled[1024,128] @ W[128,4] + b via
// V_WMMA_F32_16X16X4_F32. One wave per 16-row tile, N padded 4 -> 16,
// K consumed 4 per WMMA (32 WMMAs per wave).
// B-operand padding uses clamped index + float mask => no exec divergence:
// EXEC stays all-ones around every WMMA and loads are unconditional.
// ---------------------------------------------------------------------------
__global__ void out_gemm_wmma(const float* __restrict__ pooled,
                              const float* __restrict__ W,
                              const float* __restrict__ bias,
                              float* __restrict__ out) {
    int lane = threadIdx.x;             // 0..31, one wave per block
    int m0   = blockIdx.x * 16;
    int n    = lane & 15;               // output column (valid if < NCLASS)
    int kOff = (lane >> 4) * 2;         // 0 for lanes 0-15, 2 for lanes 16-31
    int r    = m0 + (lane & 15);        // A row owned by this lane

    int   nc   = (n < NCLASS) ? n : 0;  // clamped column (safe address)
    float mask = (n < NCLASS) ? 1.f : 0.f;
    const float* Wl = W + nc;           // per-lane column pointer

    v8f c = {};
    for (int k = 0; k < F_NODE; k += 4) {
        v2f a, b;
        a.x = pooled[(long)r * F_NODE + k + kOff];
        a.y = pooled[(long)r * F_NODE + k + kOff + 1];
        b.x = Wl[(k + kOff)     * NCLASS] * mask;
        b.y = Wl[(k + kOff + 1) * NCLASS] * mask;
        // 8 args: (neg_a, A, neg_b, B, c_mod, C, reuse_a, reuse_b)
        c = __builtin_amdgcn_wmma_f32_16x16x4_f32(
                false, a, false, b, (short)0, c, false, false);
    }

    if (n < NCLASS) {
        int rowBase = m0 + (lane >> 4) * 8;  // lanes 0-15: rows 0-7; 16-31: 8-15
        float bn = bias[n];
#pragma unroll
        for (int j = 0; j < 8; ++j)
            out[(rowBase + j) * NCLASS + n] = c[j] + bn;
    }
}

// ---------------------------------------------------------------------------
// Host-side launch. Input order: x, edge_index, edge_attr, batch, node_gamma,
// node_beta, edge_gamma, edge_beta, t, W, b, num_graphs.
// edge_attr path is dead in the reference -> never touched (saves 512MB).
// ---------------------------------------------------------------------------
extern "C" void kernel_launch(void* const* d_in, const int* in_sizes, int n_in,
                              void* d_out, int out_size, void* d_ws, size_t ws_size,
                              hipStream_t stream) {
    const float* x      = (const float*)d_in[0];
    const int*   batch  = (const int*)  d_in[3];
    const float* ngamma = (const float*)d_in[4];
    const float* nbeta  = (const float*)d_in[5];
    const float* t      = (const float*)d_in[8];
    const float* W      = (const float*)d_in[9];
    const float* bias   = (const float*)d_in[10];
    int N = in_sizes[0] / F_NODE;       // 1,000,000

    // Workspace layout (floats), ~2.6 MB total
    float* ws     = (float*)d_ws;
    float* psum   = ws;                                     // STATS_BLOCKS*128
    float* psq    = psum  + STATS_BLOCKS * F_NODE;          // STATS_BLOCKS*128
    float* scale  = psq   + STATS_BLOCKS * F_NODE;          // 128
    float* shift  = scale + F_NODE;                         // 128
    float* pooled = shift + F_NODE;                         // NGRAPH*128
    int*   start  = (int*)(pooled + (long)NGRAPH * F_NODE); // NGRAPH+1

    stats_partial <<<STATS_BLOCKS, 256, 0, stream>>>(x, N, psum, psq);
    stats_finalize<<<F_NODE, 256, 0, stream>>>(psum, psq, ngamma, nbeta, N, scale, shift);
    seg_init      <<<(NGRAPH + 1 + 255) / 256, 256, 0, stream>>>(start, N);
    seg_bounds    <<<(N + 255) / 256, 256, 0, stream>>>(batch, N, start);
    graph_pool    <<<NGRAPH, 256, 0, stream>>>(x, start, scale, shift, t, pooled);
    out_gemm_wmma <<<NGRAPH / 16, 32, 0, stream>>>(pooled, W, bias, (float*)d_out);
}